// PostProcessor_80711025426580
// MI455X (gfx1250) — compile-verified
//
#include <hip/hip_runtime.h>
#include <math.h>

#define NN 512
#define NC 151
#define ND 4096
#define TOPN 300
#define LIMIT 100
#define MPAD 112            // 100 rows padded to multiple of 16
#define NTILE 7             // MPAD / 16
#define DW_CLAMP 4.135166556742356f   // log(1000/16)

#define OUT_BOX   (LIMIT * ND)            // 409600
#define OUT_SCORE (OUT_BOX + LIMIT * 4)   // 410000
#define OUT_LAB   (OUT_SCORE + LIMIT)     // 410100
#define OUT_VALID (OUT_LAB + LIMIT)       // 410200

typedef __attribute__((ext_vector_type(2))) float v2f;
typedef __attribute__((ext_vector_type(8))) float v8f;

// ---- stable bitonic sort of 512 (key ascending, index ascending on ties) ----
__device__ __forceinline__ void bitonic512(float* key, int* idx, int t) {
  for (int k = 2; k <= 512; k <<= 1) {
    for (int j = k >> 1; j > 0; j >>= 1) {
      __syncthreads();
      int p = t ^ j;
      if (p > t) {
        float k0 = key[t], k1 = key[p];
        int i0 = idx[t], i1 = idx[p];
        bool gt = (k0 > k1) || ((k0 == k1) && (i0 > i1));
        bool up = ((t & k) == 0);
        if (gt == up) { key[t] = k1; key[p] = k0; idx[t] = i1; idx[p] = i0; }
      }
    }
  }
  __syncthreads();
}

// ---- kernel 1: softmax over classes + box decode; one block per proposal ----
__global__ void k_softmax_decode(const float* __restrict__ logits,
                                 const float* __restrict__ boxreg,
                                 const float* __restrict__ props,
                                 float* __restrict__ scores_t,   // [C][N]
                                 float* __restrict__ boxes_t) {  // [C][N][4]
  int n = blockIdx.x, t = threadIdx.x;   // blockDim = 256
  __shared__ float red[256];
  __shared__ float smax, ssum;
  float v = (t < NC) ? logits[n * NC + t] : -INFINITY;
  red[t] = v; __syncthreads();
  for (int s = 128; s > 0; s >>= 1) { if (t < s) red[t] = fmaxf(red[t], red[t + s]); __syncthreads(); }
  if (t == 0) smax = red[0];
  __syncthreads();
  float e = (t < NC) ? expf(v - smax) : 0.0f;
  red[t] = e; __syncthreads();
  for (int s = 128; s > 0; s >>= 1) { if (t < s) red[t] += red[t + s]; __syncthreads(); }
  if (t == 0) ssum = red[0];
  __syncthreads();
  if (t < NC) {
    scores_t[t * NN + n] = e / ssum;
    float x1 = props[n * 4 + 0], y1 = props[n * 4 + 1];
    float x2 = props[n * 4 + 2], y2 = props[n * 4 + 3];
    float w = x2 - x1 + 1.0f, h = y2 - y1 + 1.0f;
    float cx = x1 + 0.5f * w,  cy = y1 + 0.5f * h;
    const float* r = boxreg + (n * NC + t) * 4;
    float dx = r[0] / 10.0f, dy = r[1] / 10.0f;
    float dw = fminf(r[2] / 5.0f, DW_CLAMP);
    float dh = fminf(r[3] / 5.0f, DW_CLAMP);
    float pcx = dx * w + cx, pcy = dy * h + cy;
    float pw = expf(dw) * w, ph = expf(dh) * h;
    float bx1 = fminf(fmaxf(pcx - 0.5f * pw, 0.0f), 1023.0f);
    float by1 = fminf(fmaxf(pcy - 0.5f * ph, 0.0f), 767.0f);
    float bx2 = fminf(fmaxf(pcx + 0.5f * pw - 1.0f, 0.0f), 1023.0f);
    float by2 = fminf(fmaxf(pcy + 0.5f * ph - 1.0f, 0.0f), 767.0f);
    float* b = boxes_t + (t * NN + n) * 4;
    b[0] = bx1; b[1] = by1; b[2] = bx2; b[3] = by2;
  }
}

// ---- kernel 2: greedy NMS, one block (512 threads) per class ----
__global__ void k_nms(const float* __restrict__ scores_t,
                      const float* __restrict__ boxes_t,
                      int* __restrict__ keep_t) {              // [C][N]
  int c = blockIdx.x, t = threadIdx.x;
  if (c == 0) { keep_t[t] = 0; return; }                       // keep_all[0] = False
  __shared__ float key[512]; __shared__ int sidx[512];
  __shared__ float bx1[512], by1[512], bx2[512], by2[512], bar[512];
  __shared__ int keep[512]; __shared__ int vcount;
  float sc = scores_t[c * NN + t];
  bool valid = sc > 0.05f;
  key[t] = valid ? -sc : INFINITY;
  sidx[t] = t;
  if (t == 0) vcount = 0;
  __syncthreads();
  if (valid) atomicAdd(&vcount, 1);
  bitonic512(key, sidx, t);                                    // valid first, desc score
  int oi = sidx[t];
  const float* b = boxes_t + (c * NN + oi) * 4;
  float x1 = b[0], y1 = b[1], x2 = b[2], y2 = b[3];
  bx1[t] = x1; by1[t] = y1; bx2[t] = x2; by2[t] = y2;
  float area = (x2 - x1 + 1.0f) * (y2 - y1 + 1.0f);
  bar[t] = area;
  keep[t] = (t < vcount) ? 1 : 0;
  __syncthreads();
  int V = vcount;
  for (int i = 0; i < V; ++i) {                                // sequential suppression
    if (t > i && keep[t] && keep[i]) {
      float iw = fmaxf(fminf(x2, bx2[i]) - fmaxf(x1, bx1[i]) + 1.0f, 0.0f);
      float ih = fmaxf(fminf(y2, by2[i]) - fmaxf(y1, by1[i]) + 1.0f, 0.0f);
      float inter = iw * ih;
      if (inter / (area + bar[i] - inter) > 0.5f) keep[t] = 0;
    }
    __syncthreads();
  }
  if (t == 0) {                                                // cumsum <= TOPN cap
    int run = 0;
    for (int q = 0; q < 512; ++q)
      if (keep[q]) { run++; if (run > TOPN) keep[q] = 0; }
  }
  __syncthreads();
  keep_t[c * NN + oi] = keep[t];                               // scatter to orig order
}

// ---- kernel 3: per-proposal max/argmax over classes ----
__global__ void k_argmax(const float* __restrict__ scores_t,
                         const int* __restrict__ keep_t,
                         int* __restrict__ labels_pre,
                         float* __restrict__ masked) {
  int n = blockIdx.x * blockDim.x + threadIdx.x;
  float best = -1.0f; int bi = 0;
  for (int c = 0; c < NC; ++c) {
    float v = scores_t[c * NN + n] * (float)keep_t[c * NN + n];
    if (v > best) { best = v; bi = c; }                        // first-occurrence argmax
  }
  masked[n] = (best > 0.0f) ? best : -1.0f;
  labels_pre[n] = bi;
}

// ---- kernel 4: top-100 (stable), small outputs, one-hot selection matrix ----
__global__ void k_topk(const float* __restrict__ masked,
                       const int* __restrict__ labels_pre,
                       const float* __restrict__ boxes_t,
                       float* __restrict__ S,                  // [MPAD][N]
                       float* __restrict__ out) {
  __shared__ float key[512]; __shared__ int sidx[512];
  int t = threadIdx.x;
  key[t] = -masked[t];                                         // asc(-v, idx) == topk order
  sidx[t] = t;
  bitonic512(key, sidx, t);
  for (int q = t; q < MPAD * NN; q += 512) S[q] = 0.0f;
  __syncthreads();
  if (t < LIMIT) {
    int pi = sidx[t];
    float v = -key[t];
    bool valid = v > 0.0f;
    float m = valid ? 1.0f : 0.0f;
    int lab = labels_pre[pi];
    S[t * NN + pi] = m;                                        // one-hot row
    const float* bb = boxes_t + (lab * NN + pi) * 4;
    float* bo = out + OUT_BOX + t * 4;
    bo[0] = bb[0] * m; bo[1] = bb[1] * m; bo[2] = bb[2] * m; bo[3] = bb[3] * m;
    out[OUT_SCORE + t] = valid ? v : 0.0f;
    out[OUT_LAB + t]   = valid ? (float)lab : 0.0f;
    out[OUT_VALID + t] = m;
  }
}

// ---- kernel 5: feat_out = S @ features via V_WMMA_F32_16X16X4_F32 (exact f32) ----
// One block = 64 feature columns (4 waves x 16 cols). Each wave loads ONE B
// fragment per K-step and feeds all 7 M-tile accumulators -> features read
// exactly once (8 MB), 7 WMMAs per B-fragment load.
__global__ void __launch_bounds__(128) k_featmm(const float* __restrict__ S,
                                                const float* __restrict__ F,
                                                float* __restrict__ out) {
  int lane = threadIdx.x & 31;
  int wave = threadIdx.x >> 5;                                 // 0..3
  int ncol = blockIdx.x * 64 + wave * 16 + (lane & 15);        // feature column
  int ml = lane & 15;                                          // A row within tile
  int kk = (lane >> 4) * 2;                                    // A/B: K = vgpr + 2*(lane>=16)
  v8f acc[NTILE];
  #pragma unroll
  for (int m = 0; m < NTILE; ++m) acc[m] = (v8f){0.f,0.f,0.f,0.f,0.f,0.f,0.f,0.f};
  for (int k0 = 0; k0 < NN; k0 += 4) {
    v2f b;
    b.x = F[(k0 + kk) * ND + ncol];
    b.y = F[(k0 + kk + 1) * ND + ncol];
    #pragma unroll
    for (int m = 0; m < NTILE; ++m) {
      v2f a;
      a.x = S[(m * 16 + ml) * NN + k0 + kk];
      a.y = S[(m * 16 + ml) * NN + k0 + kk + 1];
      acc[m] = __builtin_amdgcn_wmma_f32_16x16x4_f32(false, a, false, b,
                                                     (short)0, acc[m], false, false);
    }
  }
  // D layout: row = m*16 + r + 8*(lane>=16), col = ncol. Tiles 0..5 full.
  #pragma unroll
  for (int m = 0; m < NTILE - 1; ++m) {
    int rbase = m * 16 + 8 * (lane >> 4);
    #pragma unroll
    for (int r = 0; r < 8; ++r)
      out[(rbase + r) * ND + ncol] = acc[m][r];
  }
  {                                                            // tile 6: rows 96..111
    int rbase = 96 + 8 * (lane >> 4);
    #pragma unroll
    for (int r = 0; r < 8; ++r) {
      int row = rbase + r;
      if (row < LIMIT) out[row * ND + ncol] = acc[NTILE - 1][r];
    }
  }
}

extern "C" void kernel_launch(void* const* d_in, const int* in_sizes, int n_in,
                              void* d_out, int out_size, void* d_ws, size_t ws_size,
                              hipStream_t stream) {
  const float* features = (const float*)d_in[0];   // [512][4096]
  const float* logits   = (const float*)d_in[1];   // [512][151]
  const float* boxreg   = (const float*)d_in[2];   // [512][604]
  const float* props    = (const float*)d_in[3];   // [512][4]
  float* out = (float*)d_out;

  float* scores_t   = (float*)d_ws;                         // 151*512 f32
  float* boxes_t    = scores_t + NC * NN;                   // 151*512*4 f32
  int*   keep_t     = (int*)(boxes_t + NC * NN * 4);        // 151*512 i32
  int*   labels_pre = keep_t + NC * NN;                     // 512 i32
  float* masked     = (float*)(labels_pre + NN);            // 512 f32
  float* S          = masked + NN;                          // 112*512 f32

  k_softmax_decode<<<NN, 256, 0, stream>>>(logits, boxreg, props, scores_t, boxes_t);
  k_nms<<<NC, 512, 0, stream>>>(scores_t, boxes_t, keep_t);
  k_argmax<<<2, 256, 0, stream>>>(scores_t, keep_t, labels_pre, masked);
  k_topk<<<1, 512, 0, stream>>>(masked, labels_pre, boxes_t, S, out);
  k_featmm<<<ND / 64, 128, 0, stream>>>(S, features, out);
}